// PopRSNN_33174327394835
// MI455X (gfx1250) — compile-verified
//
#include <hip/hip_runtime.h>
#include <hip/hip_bf16.h>

// ---------------------------------------------------------------------------
// PopRSNN for MI455X (gfx1250, wave32).
//   Stage A: drive GEMM (64000x300)@(300x24) via V_WMMA_F32_16X16X4_F32,
//            epilogue folds offset/v_rest/(1-decay)/noise -> aff[t,b,l].
//   Stage B: sequential leaky-integrator scan, 1 workgroup, LDS ring buffer.
//   Stage C: per-(t,area) projection (64x4)@(4x200): exactly one K=4 fp32
//            WMMA per 16x16 tile + sigmoid, writes the 307MB output
//            (the true bandwidth bottleneck: ~18us HBM floor at 23.3TB/s).
// Scratch (d_ws): aff (T*B*L floats) + rates (T*B*L floats) = 12.3 MB.
// ---------------------------------------------------------------------------

#define T_   1000
#define B_   64
#define IN_  300
#define A_   6
#define P_   4
#define L_   24      // A_*P_
#define O_   200
#define ND_  5
#define SIGMA_ 0.1f

typedef __attribute__((ext_vector_type(2))) float v2f;
typedef __attribute__((ext_vector_type(8))) float v8f;

__device__ __forceinline__ float sigmoid_f(float v) {
    return 1.0f / (1.0f + __expf(-v));
}

// ---------------------------------------------------------------------------
// Kernel A: drive.  One wave per 16-row M-tile over T*B rows.
// A-frag layout (16x4 f32): lane = 16*(k>=2) + m ; vgpr j holds K = 2*(lane>>4)+j
// B-frag layout (4x16 f32): lane = 16*(k>=2) + n ; vgpr j holds K = 2*(lane>>4)+j
// ---------------------------------------------------------------------------
__global__ void __launch_bounds__(256)
poprsnn_drive_kernel(const float* __restrict__ x,
                     const float* __restrict__ mem_noise,
                     const float* __restrict__ trial_noise,
                     const float* __restrict__ w_in,
                     const float* __restrict__ v_rest,
                     const float* __restrict__ bias_noise,
                     const float* __restrict__ trial_offset_w,
                     const float* __restrict__ decay_v,
                     float* __restrict__ aff)
{
    const int lane   = threadIdx.x & 31;
    const int wave   = threadIdx.x >> 5;
    const int waveId = blockIdx.x * (blockDim.x >> 5) + wave;
    const int mbase  = waveId * 16;              // row tile over T*B = 64000
    const int m      = lane & 15;
    const int khalf  = lane >> 4;
    const int row    = mbase + m;

    // B fragment column indices (masked arithmetically: EXEC stays all-ones)
    const int   n1  = lane & 15;                 // N-tile 0: l = 0..15
    const int   n2  = 16 + (lane & 15);          // N-tile 1: l = 16..31 (valid < 24)
    const float bs2 = (n2 < L_) ? 1.0f : 0.0f;
    const int   n2c = (n2 < L_) ? n2 : (L_ - 1);

    v8f c0 = {};
    v8f c1 = {};

    const float* xrow = x + (long)row * IN_ + khalf * 2;
    const float* wr1  = w_in + (long)n1  * IN_ + khalf * 2;
    const float* wr2  = w_in + (long)n2c * IN_ + khalf * 2;

    #pragma unroll 5
    for (int kk = 0; kk < IN_ / 4; ++kk) {
        const int k4 = kk * 4;
        v2f a;  a.x  = xrow[k4];     a.y  = xrow[k4 + 1];
        v2f b0; b0.x = wr1[k4];      b0.y = wr1[k4 + 1];
        v2f b1; b1.x = wr2[k4]*bs2;  b1.y = wr2[k4 + 1]*bs2;
        c0 = __builtin_amdgcn_wmma_f32_16x16x4_f32(false, a, false, b0,
                                                   (short)0, c0, false, false);
        c1 = __builtin_amdgcn_wmma_f32_16x16x4_f32(false, a, false, b1,
                                                   (short)0, c1, false, false);
    }

    // Epilogue: drive + trial offset + v_rest, then fold (1-decay) and noise.
    // C/D layout: lanes 0-15 -> M=r ; lanes 16-31 -> M=8+r ; N = lane&15.
    const int nOut = lane & 15;
    const int mOff = khalf * 8;
    #pragma unroll
    for (int r = 0; r < 8; ++r) {
        const int orow = mbase + mOff + r;       // row in [0, T*B)
        const int bidx = orow & (B_ - 1);        // B_ == 64 is a power of 2
        const float tn0 = trial_noise[bidx * 2 + 0];
        const float tn1 = trial_noise[bidx * 2 + 1];
        {   // N-tile 0: l = nOut
            const int l = nOut;
            float val = c0[r] + tn0 * trial_offset_w[l]
                              + tn1 * trial_offset_w[L_ + l] + v_rest[l];
            float av  = (1.0f - decay_v[l]) * val
                      + SIGMA_ * bias_noise[l] * mem_noise[(long)orow * L_ + l];
            aff[(long)orow * L_ + l] = av;
        }
        const int l2 = 16 + nOut;                // N-tile 1 (valid cols only)
        if (l2 < L_) {
            float val = c1[r] + tn0 * trial_offset_w[l2]
                              + tn1 * trial_offset_w[L_ + l2] + v_rest[l2];
            float av  = (1.0f - decay_v[l2]) * val
                      + SIGMA_ * bias_noise[l2] * mem_noise[(long)orow * L_ + l2];
            aff[(long)orow * L_ + l2] = av;
        }
    }
}

// ---------------------------------------------------------------------------
// Kernel B: sequential scan, one workgroup. 512 threads x 3 state elems each.
// ring[slot][b][l] holds rate[t-ND .. t-1]; slot t%ND is rate[t-ND] at read.
// v = decay*v + (1-decay)*i_rec + aff   (aff already contains the rest)
// ---------------------------------------------------------------------------
__global__ void __launch_bounds__(512)
poprsnn_scan_kernel(const float* __restrict__ w_rec,      // (1,L,L): w[l*L+j]
                    const float* __restrict__ decay_v,
                    const float* __restrict__ aff,
                    float* __restrict__ rates)
{
    __shared__ float wrec_s[L_ * L_];
    __shared__ float ring[ND_][B_][L_];

    const int tid = threadIdx.x;
    for (int i = tid; i < L_ * L_; i += 512) wrec_s[i] = w_rec[i];
    for (int i = tid; i < ND_ * B_ * L_; i += 512) (&ring[0][0][0])[i] = 0.0f;
    __syncthreads();

    float v[3]; int bi[3], li[3]; float dec[3];
    #pragma unroll
    for (int r = 0; r < 3; ++r) {
        const int e = tid + 512 * r;             // e < 1536 = B_*L_
        bi[r] = e / L_;  li[r] = e % L_;
        v[r] = 0.0f;     dec[r] = decay_v[li[r]];
    }

    for (int t = 0; t < T_; ++t) {
        const int slot = t % ND_;
        float rnew[3];
        #pragma unroll
        for (int r = 0; r < 3; ++r) {
            const int b = bi[r], l = li[r];
            float acc = 0.0f;
            #pragma unroll
            for (int j = 0; j < L_; ++j)
                acc += ring[slot][b][j] * wrec_s[l * L_ + j];
            const float d  = dec[r];
            const float vv = d * v[r] + (1.0f - d) * acc
                           + aff[((long)t * B_ + b) * L_ + l];
            v[r]    = vv;
            rnew[r] = sigmoid_f(vv);
        }
        __syncthreads();                          // all reads of slot done
        #pragma unroll
        for (int r = 0; r < 3; ++r) {
            ring[slot][bi[r]][li[r]] = rnew[r];
            rates[((long)t * B_ + bi[r]) * L_ + li[r]] = rnew[r];
        }
        __syncthreads();                          // slot updated for t+1
    }
}

// ---------------------------------------------------------------------------
// Kernel C: projection. grid = (13 n-tiles, 6 areas, 1000 t), 4 waves/block,
// each wave = one batch M-tile (16 rows). K = P_ = 4 -> exactly one fp32 WMMA.
// ---------------------------------------------------------------------------
__global__ void __launch_bounds__(128)
poprsnn_proj_kernel(const float* __restrict__ rates,
                    const float* __restrict__ proj_w,     // (A,O,P)
                    const float* __restrict__ proj_b,     // (A,O)
                    float* __restrict__ out)              // (T,B,A*O)
{
    const int t     = blockIdx.z;
    const int area  = blockIdx.y;
    const int ntile = blockIdx.x;                 // 0..12, O = 12*16 + 8
    const int wave  = threadIdx.x >> 5;
    const int lane  = threadIdx.x & 31;
    const int mbase = wave * 16;                  // batch tile base
    const int khalf = lane >> 4;

    // A fragment: pop rates (16 batches x 4 pops)
    const int b = mbase + (lane & 15);
    const float* rrow = rates + ((long)t * B_ + b) * L_ + area * P_ + khalf * 2;
    v2f a;  a.x = rrow[0];  a.y = rrow[1];

    // B fragment: proj_w[area][o][p], o = ntile*16 + n (masked arithmetically)
    const int   o  = ntile * 16 + (lane & 15);
    const int   oc = (o < O_) ? o : (O_ - 1);
    const float bs = (o < O_) ? 1.0f : 0.0f;
    const float* wrow = proj_w + ((long)area * O_ + oc) * P_ + khalf * 2;
    v2f bf;  bf.x = wrow[0] * bs;  bf.y = wrow[1] * bs;

    v8f c = {};
    c = __builtin_amdgcn_wmma_f32_16x16x4_f32(false, a, false, bf,
                                              (short)0, c, false, false);

    // Epilogue: + bias, sigmoid, store.  lanes 0-15 -> M=r ; 16-31 -> M=8+r.
    const int oo = ntile * 16 + (lane & 15);
    if (oo < O_) {
        const float bias = proj_b[area * O_ + oo];
        const int mOff = khalf * 8;
        #pragma unroll
        for (int r = 0; r < 8; ++r) {
            const int bb = mbase + mOff + r;
            const float val = sigmoid_f(c[r] + bias);
            out[((size_t)t * B_ + bb) * (A_ * O_) + area * O_ + oo] = val;
        }
    }
}

// ---------------------------------------------------------------------------
extern "C" void kernel_launch(void* const* d_in, const int* in_sizes, int n_in,
                              void* d_out, int out_size, void* d_ws, size_t ws_size,
                              hipStream_t stream) {
    const float* x              = (const float*)d_in[0];
    const float* mem_noise      = (const float*)d_in[1];
    const float* trial_noise    = (const float*)d_in[2];
    const float* w_in           = (const float*)d_in[3];
    const float* w_rec          = (const float*)d_in[4];
    const float* v_rest         = (const float*)d_in[5];
    const float* bias_noise     = (const float*)d_in[6];
    const float* trial_offset_w = (const float*)d_in[7];
    const float* decay_v        = (const float*)d_in[8];
    const float* proj_w         = (const float*)d_in[9];
    const float* proj_b         = (const float*)d_in[10];
    float* out = (float*)d_out;

    // Scratch layout: aff | rates, each T*B*L floats (6.14 MB each).
    float* aff   = (float*)d_ws;
    float* rates = aff + (size_t)T_ * B_ * L_;

    // Stage A: 64000 rows / 16 = 4000 waves ; 8 waves (256 thr) per block.
    poprsnn_drive_kernel<<<4000 / 8, 256, 0, stream>>>(
        x, mem_noise, trial_noise, w_in, v_rest, bias_noise,
        trial_offset_w, decay_v, aff);

    // Stage B: single-workgroup sequential scan.
    poprsnn_scan_kernel<<<1, 512, 0, stream>>>(w_rec, decay_v, aff, rates);

    // Stage C: (13 n-tiles, 6 areas, 1000 timesteps), 4 waves per block.
    poprsnn_proj_kernel<<<dim3(13, A_, T_), 128, 0, stream>>>(
        rates, proj_w, proj_b, out);
}